// AutoCorrelation_1571958031021
// MI455X (gfx1250) — compile-verified
//
#include <hip/hip_runtime.h>
#include <hip/hip_bf16.h>

typedef __attribute__((ext_vector_type(2))) float v2f;
typedef __attribute__((ext_vector_type(8))) float v8f;

#define S_LEN   4096
#define D_DIM   64
#define NTHR    256

// ---------------------------------------------------------------------------
// Kernel 1: per (b,h,d) column
//   corr[tau] = sum_t q[(t+tau)%S]*k[t]  -> bitonic sort desc -> softmax
//   writes W into d_out at [bh, i, d]  (same layout as final output)
// ---------------------------------------------------------------------------
__global__ __launch_bounds__(NTHR) void corr_sort_softmax_kernel(
    const float* __restrict__ q, const float* __restrict__ k,
    float* __restrict__ wout)
{
    __shared__ float qs[2 * S_LEN];   // duplicated q: no modulo in inner loop
    __shared__ float ks[S_LEN];       // k column; reused as reduction scratch
    __shared__ float corr[S_LEN];

    const int tid = threadIdx.x;
    const int bhd = blockIdx.x;              // 0 .. B*H*D-1
    const int d   = bhd & (D_DIM - 1);
    const int bh  = bhd >> 6;                // D_DIM == 64
    const size_t base = (size_t)bh * S_LEN * D_DIM + d;

    // strided column gather (L2-resident after first touch)
    for (int t = tid; t < S_LEN; t += NTHR) {
        float qv = q[base + (size_t)t * D_DIM];
        float kv = k[base + (size_t)t * D_DIM];
        qs[t]         = qv;
        qs[t + S_LEN] = qv;
        ks[t]         = kv;
    }
    __syncthreads();

    // --- correlation: thread owns tau = tid*16 .. tid*16+15 ---
    float acc[16];
#pragma unroll
    for (int j = 0; j < 16; ++j) acc[j] = 0.f;
    const int t0 = tid * 16;

    for (int t = 0; t < S_LEN; t += 4) {
        float4 kv = *reinterpret_cast<const float4*>(&ks[t]);
        const float* qp = &qs[t + t0];          // float4-aligned (t%4==0, t0%16==0)
        float w[20];
#pragma unroll
        for (int u = 0; u < 5; ++u) {
            float4 qv = *reinterpret_cast<const float4*>(qp + 4 * u);
            w[4*u+0] = qv.x; w[4*u+1] = qv.y; w[4*u+2] = qv.z; w[4*u+3] = qv.w;
        }
#pragma unroll
        for (int j = 0; j < 16; ++j) {
            acc[j] = fmaf(w[j    ], kv.x, acc[j]);
            acc[j] = fmaf(w[j + 1], kv.y, acc[j]);
            acc[j] = fmaf(w[j + 2], kv.z, acc[j]);
            acc[j] = fmaf(w[j + 3], kv.w, acc[j]);
        }
    }
#pragma unroll
    for (int j = 0; j < 16; ++j) corr[t0 + j] = acc[j];
    __syncthreads();

    // --- bitonic sort, descending ---
    for (int kk = 2; kk <= S_LEN; kk <<= 1) {
        for (int j = kk >> 1; j > 0; j >>= 1) {
            for (int i = tid; i < S_LEN; i += NTHR) {
                int l = i ^ j;
                if (l > i) {
                    float a = corr[i], b = corr[l];
                    bool up = (i & kk) == 0;          // descending network
                    bool sw = up ? (a < b) : (a > b);
                    if (sw) { corr[i] = b; corr[l] = a; }
                }
            }
            __syncthreads();
        }
    }

    // --- softmax over sorted values (max == corr[0]) ---
    const float m = corr[0];
    __syncthreads();
    float psum = 0.f;
    for (int i = tid; i < S_LEN; i += NTHR) {
        float e = expf(corr[i] - m);
        corr[i] = e;
        psum += e;
    }
    ks[tid] = psum;                 // k column dead; reuse as scratch
    __syncthreads();
    for (int s = NTHR >> 1; s > 0; s >>= 1) {
        if (tid < s) ks[tid] += ks[tid + s];
        __syncthreads();
    }
    const float inv = 1.0f / ks[0];
    for (int i = tid; i < S_LEN; i += NTHR)
        wout[(size_t)bh * S_LEN * D_DIM + (size_t)i * D_DIM + d] = corr[i] * inv;
}

// ---------------------------------------------------------------------------
// Kernel 2: per (b,h):  out[4096,64] = W[4096,64] @ V[0:64,0:64]  (in place)
// V_WMMA_F32_16X16X4_F32: A 16x4, B 4x16, C/D 16x16 f32.
//   A: lane(0-15)=M, v0/v1 = K {0,1}; lanes 16-31: K {2,3}
//   B: lane(0-15)=N, v0/v1 = K {0,1}; lanes 16-31: K {2,3}
//   D: vgpr j -> M = j + 8*(lane>=16), N = lane&15
// 4 waves per block, each owns a disjoint 16-row block -> in-place safe.
// ---------------------------------------------------------------------------
__global__ __launch_bounds__(128) void wmma_apply_kernel(
    const float* __restrict__ vals, float* out)
{
    __shared__ float vsh[D_DIM * D_DIM];     // 16 KB: V head, row-major [j][l]

    const int tid = threadIdx.x;
    const int bh  = blockIdx.x >> 6;         // 64 row-blocks of 64 rows per bh
    const int rb  = blockIdx.x & 63;
    const size_t sbase = (size_t)bh * S_LEN * D_DIM;

    for (int i = tid; i < D_DIM * D_DIM; i += 128)
        vsh[i] = vals[sbase + i];            // values[b,h,j<64,l]
    __syncthreads();

    const int lane = tid & 31;
    const int wave = tid >> 5;
    const int r0   = rb * 64 + wave * 16;
    const int m    = lane & 15;
    const int kh   = lane >> 4;              // lane-half selects K {0,1} vs {2,3}

    float* W = out + sbase;                  // W lives in d_out (in-place)
    v8f acc0 = {}, acc1 = {}, acc2 = {}, acc3 = {};

#pragma unroll
    for (int kk = 0; kk < 16; ++kk) {
        const int ka = 4 * kk + 2 * kh;
        const float* ap = W + (size_t)(r0 + m) * D_DIM + ka;
        v2f a; a.x = ap[0]; a.y = ap[1];

        v2f b0, b1, b2, b3;
        b0.x = vsh[ka * D_DIM +      m]; b0.y = vsh[(ka + 1) * D_DIM +      m];
        b1.x = vsh[ka * D_DIM + 16 + m]; b1.y = vsh[(ka + 1) * D_DIM + 16 + m];
        b2.x = vsh[ka * D_DIM + 32 + m]; b2.y = vsh[(ka + 1) * D_DIM + 32 + m];
        b3.x = vsh[ka * D_DIM + 48 + m]; b3.y = vsh[(ka + 1) * D_DIM + 48 + m];

        acc0 = __builtin_amdgcn_wmma_f32_16x16x4_f32(false, a, false, b0, (short)0, acc0, false, false);
        acc1 = __builtin_amdgcn_wmma_f32_16x16x4_f32(false, a, false, b1, (short)0, acc1, false, false);
        acc2 = __builtin_amdgcn_wmma_f32_16x16x4_f32(false, a, false, b2, (short)0, acc2, false, false);
        acc3 = __builtin_amdgcn_wmma_f32_16x16x4_f32(false, a, false, b3, (short)0, acc3, false, false);
    }

#pragma unroll
    for (int j = 0; j < 8; ++j) {
        const int row = r0 + 8 * kh + j;
        float* op = W + (size_t)row * D_DIM + m;
        op[0]  = acc0[j];
        op[16] = acc1[j];
        op[32] = acc2[j];
        op[48] = acc3[j];
    }
}

// ---------------------------------------------------------------------------
extern "C" void kernel_launch(void* const* d_in, const int* in_sizes, int n_in,
                              void* d_out, int out_size, void* d_ws, size_t ws_size,
                              hipStream_t stream) {
    const float* q = (const float*)d_in[0];
    const float* k = (const float*)d_in[1];
    const float* v = (const float*)d_in[2];
    float* out = (float*)d_out;

    const int BH = in_sizes[0] / (S_LEN * D_DIM);   // B*H = 64

    corr_sort_softmax_kernel<<<BH * D_DIM, NTHR, 0, stream>>>(q, k, out);
    wmma_apply_kernel<<<BH * (S_LEN / 64), 128, 0, stream>>>(v, out);
}